// TestSeqNMSListModule_32779190403244
// MI455X (gfx1250) — compile-verified
//
#include <hip/hip_runtime.h>
#include <stdint.h>

#define F_FRAMES 16
#define N_BOX    256
#define NW       8          // 256 bits / 32 = 8 mask words per row
#define IOU_THF  0.2f
#define NEGV     (-1e30f)
#define MAX_IT   1024

typedef float    v2f  __attribute__((ext_vector_type(2)));
typedef float    v8f  __attribute__((ext_vector_type(8)));
typedef uint32_t u32x4 __attribute__((ext_vector_type(4)));
typedef uint32_t u32x8 __attribute__((ext_vector_type(8)));

// ---- LDS layout (byte offsets inside dynamic shared region) ----
// intra mask : 16*256*8*4  = 131072
// link  mask : 15*256*8*4  = 122880
// temp region (64KB): boxes during mask build, then recycled:
//   scores 16KB | dp 16KB | nxt 8KB | cont 4KB | aux
#define OFF_INTRA   0u
#define OFF_LINK    131072u
#define OFF_TEMP    253952u
#define OFF_SCORES  253952u
#define OFF_DP      270336u
#define OFF_NXT     286720u
#define OFF_CONT    294912u
#define OFF_AUX     299008u
#define SMEM_BYTES  319488u

__device__ inline bool iou_ge(const float4 a, const float4 b, float th) {
  float x1 = fmaxf(a.x, b.x), y1 = fmaxf(a.y, b.y);
  float x2 = fminf(a.z, b.z), y2 = fminf(a.w, b.w);
  float inter = fmaxf(x2 - x1, 0.0f) * fmaxf(y2 - y1, 0.0f);
  float aa = (a.z - a.x) * (a.w - a.y);
  float ab = (b.z - b.x) * (b.w - b.y);
  float uni = fmaxf(aa + ab - inter, 1e-8f);
  return (inter / uni) >= th;
}

// Tensor Data Mover: 1-D contiguous copy of nelem 4-byte elements into LDS.
// D# per CDNA5 ISA: group0 = {flags, lds_addr, gaddr_lo, gaddr_hi|type},
// group1 = {data_size=4B, tensor_dim0=nelem, tensor_dim1=1, tile_dim0=nelem, tile_dim1=1}.
__device__ inline void tdm_load_1d(uint32_t lds_byte, uint64_t gaddr, uint32_t nelem) {
  u32x4 g0;
  g0[0] = 1u;                                                 // count=1, user mode
  g0[1] = lds_byte;                                           // lds_addr
  g0[2] = (uint32_t)(gaddr & 0xffffffffull);                  // global_addr[31:0]
  g0[3] = (uint32_t)((gaddr >> 32) & 0x1ffffffull) | (2u << 30); // addr[56:32], type=2
  u32x8 g1;
  g1[0] = (2u << 16);                 // workgroup_mask=0, data_size=2 (4 bytes)
  g1[1] = (nelem & 0xffffu) << 16;    // tensor_dim0[15:0]
  g1[2] = (nelem >> 16) | (1u << 16); // tensor_dim0[31:16], tensor_dim1 lo = 1
  g1[3] = (nelem << 16);              // tensor_dim1 hi = 0, tile_dim0 = nelem (<=65535)
  g1[4] = 1u;                         // tile_dim1 = 1, tile_dim2 = 0
  g1[5] = nelem;                      // tensor_dim0_stride lo
  g1[6] = 0u;
  g1[7] = 0u;
  asm volatile("tensor_load_to_lds %0, %1" :: "s"(g0), "s"(g1) : "memory");
}

extern "C" __global__ __launch_bounds__(1024)
void seqnms_gfx1250(const float* __restrict__ boxes_g,
                    const float* __restrict__ scores_g,
                    const int*   __restrict__ classes_g,
                    float* __restrict__ out) {
  extern __shared__ char smem[];
  const uint32_t lds_base = (uint32_t)(uintptr_t)smem;  // low 32 bits = LDS offset

  uint32_t* intra    = (uint32_t*)(smem + OFF_INTRA);
  uint32_t* linkm    = (uint32_t*)(smem + OFF_LINK);
  float4*   boxes_s  = (float4*)  (smem + OFF_TEMP);
  float*    scores_s = (float*)   (smem + OFF_SCORES);
  float*    dp       = (float*)   (smem + OFF_DP);
  short*    nxt      = (short*)   (smem + OFF_NXT);
  char*     cont     = (char*)    (smem + OFF_CONT);
  uint32_t* alive    = (uint32_t*)(smem + OFF_AUX);          // 16*8 words
  float*    wval     = (float*)   (smem + OFF_AUX + 512);    // 32
  int*      widx     = (int*)     (smem + OFF_AUX + 640);    // 32
  int*      memb     = (int*)     (smem + OFF_AUX + 768);    // 16
  float*    auxf     = (float*)   (smem + OFF_AUX + 832);    // [0]=avg
  int*      auxi     = (int*)     (smem + OFF_AUX + 836);    // [0]=done

  const int tid = threadIdx.x;

  // ---- Phase A: TDM stage boxes (64KB) into LDS (wave 0 only) ----
  if (tid < 32) {
    tdm_load_1d(lds_base + OFF_TEMP, (uint64_t)(uintptr_t)boxes_g, F_FRAMES * N_BOX * 4);
    __builtin_amdgcn_s_wait_tensorcnt(0);
  }
  __syncthreads();

  // ---- Phase B: build intra (suppression) and link (same-class linkage) bitmasks ----
  for (int idx = tid; idx < F_FRAMES * N_BOX * NW; idx += 1024) {
    int f = idx >> 11, rem = idx & 2047, i = rem >> 3, w = rem & 7;
    float4 a = boxes_s[f * N_BOX + i];
    uint32_t bits = 0u;
    for (int b = 0; b < 32; ++b)
      if (iou_ge(a, boxes_s[f * N_BOX + (w << 5) + b], IOU_THF)) bits |= (1u << b);
    intra[idx] = bits;
  }
  for (int idx = tid; idx < (F_FRAMES - 1) * N_BOX * NW; idx += 1024) {
    int f = idx >> 11, rem = idx & 2047, i = rem >> 3, w = rem & 7;
    float4 a = boxes_s[f * N_BOX + i];
    int ca = classes_g[f * N_BOX + i];
    uint32_t bits = 0u;
    for (int b = 0; b < 32; ++b) {
      int j = (w << 5) + b;
      if (classes_g[(f + 1) * N_BOX + j] == ca &&
          iou_ge(a, boxes_s[(f + 1) * N_BOX + j], IOU_THF))
        bits |= (1u << b);
    }
    linkm[idx] = bits;
  }
  __syncthreads();  // boxes region dead after this point

  // ---- Phase C: TDM stage scores (recycled region) + init alive/done ----
  if (tid < 32) {
    tdm_load_1d(lds_base + OFF_SCORES, (uint64_t)(uintptr_t)scores_g, F_FRAMES * N_BOX);
    __builtin_amdgcn_s_wait_tensorcnt(0);
  }
  if (tid >= 32 && tid < 32 + F_FRAMES * NW) alive[tid - 32] = 0xffffffffu;
  if (tid == 200) auxi[0] = 0;
  __syncthreads();

  // ---- Phase D: Seq-NMS iterations (done-gated like the reference scan) ----
  for (int it = 0; it < MAX_IT; ++it) {
    // dp[15] = s_eff[15]
    if (tid < N_BOX) {
      int i = tid;
      bool al = (alive[15 * NW + (i >> 5)] >> (i & 31)) & 1u;
      dp[15 * N_BOX + i]   = al ? scores_s[15 * N_BOX + i] : NEGV;
      cont[15 * N_BOX + i] = 0;
      nxt[15 * N_BOX + i]  = 0;
    }
    __syncthreads();

    // backward DP: 4 lanes per row, bit-scan over link & alive
    for (int f = 14; f >= 0; --f) {
      int i = tid >> 2, sub = tid & 3;
      float mx = -3.0e38f; int arg = 0;
      for (int ww = 0; ww < 2; ++ww) {
        int w = sub * 2 + ww;
        uint32_t m = linkm[(f * N_BOX + i) * NW + w] & alive[(f + 1) * NW + w];
        while (m) {
          int b = __builtin_ctz(m); m &= m - 1;
          int j = (w << 5) + b;
          float v = dp[(f + 1) * N_BOX + j];
          if (v > mx) { mx = v; arg = j; }   // ascending scan => first-max kept
        }
      }
      for (int off = 1; off < 4; off <<= 1) {
        float ov = __shfl_xor(mx, off, 4);
        int   oa = __shfl_xor(arg, off, 4);
        if (ov > mx || (ov == mx && oa < arg)) { mx = ov; arg = oa; }
      }
      if (sub == 0) {
        bool al = (alive[f * NW + (i >> 5)] >> (i & 31)) & 1u;
        float s = al ? scores_s[f * N_BOX + i] : NEGV;
        bool c = mx > 0.0f;
        dp[f * N_BOX + i]   = s + (c ? mx : 0.0f);
        cont[f * N_BOX + i] = c ? 1 : 0;
        nxt[f * N_BOX + i]  = (short)arg;
      }
      __syncthreads();
    }

    // global argmax over dp[16*256] (first-occurrence tie-break)
    {
      float bv = -3.4e38f; int bi = 0;
      int base = tid * 4;
      for (int k = 0; k < 4; ++k) {
        float v = dp[base + k];
        if (v > bv || (v == bv && (base + k) < bi)) { bv = v; bi = base + k; }
      }
      for (int off = 16; off >= 1; off >>= 1) {
        float ov = __shfl_xor(bv, off, 32);
        int   oi = __shfl_xor(bi, off, 32);
        if (ov > bv || (ov == bv && oi < bi)) { bv = ov; bi = oi; }
      }
      if ((tid & 31) == 0) { wval[tid >> 5] = bv; widx[tid >> 5] = bi; }
    }
    __syncthreads();

    // path walk / rescore decision (serial, tiny: <=16 steps)
    if (tid == 0) {
      float bv = wval[0]; int bi = widx[0];
      for (int w = 1; w < 32; ++w)
        if (wval[w] > bv || (wval[w] == bv && widx[w] < bi)) { bv = wval[w]; bi = widx[w]; }
      int fs = bi >> 8, i0 = bi & 255;
      for (int f = 0; f < F_FRAMES; ++f) memb[f] = -1;
      int cur = i0, len = 0; float ssum = 0.0f;
      for (int f = fs; f < F_FRAMES; ++f) {
        memb[f] = cur; ++len; ssum += scores_s[f * N_BOX + cur];
        if (!cont[f * N_BOX + cur]) break;
        cur = (int)nxt[f * N_BOX + cur];
      }
      if (len < 2) auxi[0] = 1;                 // done: all later iterations are no-ops
      else         auxf[0] = ssum / (float)len; // mean rescore
    }
    __syncthreads();
    if (auxi[0]) break;

    // rescore members; suppress alive bits by the member's intra-IoU row
    if (tid < F_FRAMES) {
      int m = memb[tid];
      if (m >= 0) scores_s[tid * N_BOX + m] = auxf[0];
    }
    if (tid >= 64 && tid < 64 + F_FRAMES * NW) {
      int t = tid - 64, f = t >> 3, w = t & 7;
      int m = memb[f];
      if (m >= 0) alive[f * NW + w] &= ~intra[(f * N_BOX + m) * NW + w];
    }
    __syncthreads();
  }
  __syncthreads();

  // ---- Phase E: WMMA epilogue — out = I16 x scores (exact), one 16x16 tile/wave ----
  int wid = tid >> 5, lane = tid & 31;
  if (wid < 16) {
    int colBase = wid * 16;
    int half = lane >> 4, l = lane & 15;
    v8f acc = {0.f, 0.f, 0.f, 0.f, 0.f, 0.f, 0.f, 0.f};
    for (int c = 0; c < 4; ++c) {
      int k0 = 4 * c + 2 * half;                 // A: lane(M=l), vgpr v -> K = 4c + 2*half + v
      v2f a, b;
      a.x = (l == k0)     ? 1.0f : 0.0f;         // identity columns
      a.y = (l == k0 + 1) ? 1.0f : 0.0f;
      b.x = scores_s[(k0    ) * N_BOX + colBase + l];  // B: vgpr v -> row K = 4c + 2*half + v
      b.y = scores_s[(k0 + 1) * N_BOX + colBase + l];
      acc = __builtin_amdgcn_wmma_f32_16x16x4_f32(false, a, false, b, (short)0, acc,
                                                  false, false);
    }
    for (int r = 0; r < 8; ++r) {
      int row = r + 8 * half;                    // D: vgpr r -> M=r (lanes 0-15), M=r+8 (16-31)
      out[row * N_BOX + colBase + l] = acc[r];
    }
  }
}

extern "C" void kernel_launch(void* const* d_in, const int* in_sizes, int n_in,
                              void* d_out, int out_size, void* d_ws, size_t ws_size,
                              hipStream_t stream) {
  (void)in_sizes; (void)n_in; (void)d_ws; (void)ws_size; (void)out_size;
  const float* boxes   = (const float*)d_in[0];
  const float* scores  = (const float*)d_in[1];
  const int*   classes = (const int*)d_in[2];
  float* out = (float*)d_out;
  hipFuncSetAttribute((const void*)seqnms_gfx1250,
                      hipFuncAttributeMaxDynamicSharedMemorySize, (int)SMEM_BYTES);
  seqnms_gfx1250<<<dim3(1), dim3(1024), SMEM_BYTES, stream>>>(boxes, scores, classes, out);
}